// PhysicsSimulator_88742614270224
// MI455X (gfx1250) — compile-verified
//
#include <hip/hip_runtime.h>
#include <hip/hip_bf16.h>

#define NB 2048
#define DT 0.01f
#define REST 0.5f
#define STIFF 1e4f

typedef float v2f __attribute__((ext_vector_type(2)));
typedef float v8f __attribute__((ext_vector_type(8)));

// ---------------------------------------------------------------------------
// Kernel 0: zero the global force/torque accumulator (d_ws is poisoned 0xAA).
// ---------------------------------------------------------------------------
__global__ void Phys_zero_kernel(float* __restrict__ acc, int n) {
    int i = blockIdx.x * blockDim.x + threadIdx.x;
    if (i < n) acc[i] = 0.0f;
}

// ---------------------------------------------------------------------------
// Kernel 1: WMMA broadphase (dist^2 Gram matrix per 16x16 tile) + sparse
// narrowphase contact solve with LDS accumulation, global atomic merge.
// grid = (128 i-tiles, 4 j-chunks), block = 256 threads = 8 wave32.
// Wave w of chunk cy handles j-tiles {cy*32 + w + 8k, k=0..3}.
// Broadphase keeps per-element slack t[v] = rsum^2 - dist^2 in registers
// (collision <=> t > 0) and any-tests via a max tree: no sqrt, no bitmask
// packing in the hot loop. Self-pairs are filtered in the rare path.
// ---------------------------------------------------------------------------
__global__ void __launch_bounds__(256)
Phys_collide_kernel(const float* __restrict__ pos,
                    const float* __restrict__ lv,
                    const float* __restrict__ av,
                    const float* __restrict__ rad,
                    const float* __restrict__ mass,
                    float* __restrict__ gacc) {
    __shared__ float pI[16][3];
    __shared__ float lvI[16][3];
    __shared__ float wI[16][3];
    __shared__ float rI[16];
    __shared__ float mI[16];
    __shared__ float sqI[16];
    __shared__ float acc[96];   // [row][fx,fy,fz,tx,ty,tz]

    const int i0  = blockIdx.x * 16;
    const int tid = threadIdx.x;

    if (tid < 16) {
        const int a = i0 + tid;
        const float x = pos[3*a], y = pos[3*a+1], z = pos[3*a+2];
        pI[tid][0] = x; pI[tid][1] = y; pI[tid][2] = z;
        sqI[tid] = x*x + y*y + z*z;
        lvI[tid][0] = lv[3*a]; lvI[tid][1] = lv[3*a+1]; lvI[tid][2] = lv[3*a+2];
        wI[tid][0]  = av[3*a]; wI[tid][1]  = av[3*a+1]; wI[tid][2]  = av[3*a+2];
        rI[tid] = rad[a];
        mI[tid] = mass[a];
    }
    if (tid < 96) acc[tid] = 0.0f;
    __syncthreads();

    const int wave = tid >> 5;
    const int lane = tid & 31;
    const int half = lane >> 4;   // 0: K=0,1 / rows 0-7 ; 1: K=2,3 / rows 8-15
    const int col  = lane & 15;   // N (column) for B/C/D fragments, M for A

    // A fragment (16x4 f32): A[m][k] = -2 * p_i[m][k] (k<3), 0 for k==3.
    v2f Af;
    Af[0] = half ? (-2.0f * pI[col][2]) : (-2.0f * pI[col][0]);
    Af[1] = half ? 0.0f                 : (-2.0f * pI[col][1]);

    // Per-lane register copies of |p_i|^2 and r_i for the 8 rows this lane
    // owns in the C/D fragment (M = v + 8*half).
    float sqr[8], rIr[8];
#pragma unroll
    for (int v = 0; v < 8; ++v) {
        sqr[v] = sqI[v + 8 * half];
        rIr[v] = rI[v + 8 * half];
    }
    // Which v (if any) is this lane's diagonal element (m == col).
    const int vdiag = col - 8 * half;   // in [0,8) only for the owning half

    const int jt_end = blockIdx.y * 32 + 32;
    for (int jt = blockIdx.y * 32 + wave; jt < jt_end; jt += 8) {
        const int b = jt * 16 + col;
        const float bx = pos[3*b], by = pos[3*b+1], bz = pos[3*b+2];
        const float rj = rad[b];
        const float sqj = bx*bx + by*by + bz*bz;

        // B fragment (4x16 f32): B[k][n] = p_j[n][k] (k<3), 0 for k==3.
        v2f Bf;
        Bf[0] = half ? bz   : bx;
        Bf[1] = half ? 0.0f : by;

        // C = |p_i|^2 + |p_j|^2  ->  D = C - 2 p_i . p_j = dist^2.
        v8f Cf;
#pragma unroll
        for (int v = 0; v < 8; ++v) Cf[v] = sqr[v] + sqj;

        v8f Df = __builtin_amdgcn_wmma_f32_16x16x4_f32(
            /*neg_a=*/false, Af, /*neg_b=*/false, Bf,
            /*c_mod=*/(short)0, Cf, /*reuse_a=*/false, /*reuse_b=*/false);

        // Slack t = rsum^2 - dist^2; collision <=> t > 0. Any-test via max
        // tree (no bitmask packing). Diagonal tiles self-trigger but are
        // filtered below (only 1 diagonal tile per i-tile, negligible).
        float t[8];
#pragma unroll
        for (int v = 0; v < 8; ++v) {
            const float rsum = rIr[v] + rj;
            t[v] = __builtin_fmaf(rsum, rsum, -Df[v]);
        }
        float tmax = fmaxf(fmaxf(fmaxf(t[0], t[1]), fmaxf(t[2], t[3])),
                           fmaxf(fmaxf(t[4], t[5]), fmaxf(t[6], t[7])));

        if (__any((int)(tmax > 0.0f))) {
            // Narrowphase: rare path. Load column-b dynamics once per lane.
            const bool diagTile = (jt == blockIdx.x);
            const float lvbx = lv[3*b], lvby = lv[3*b+1], lvbz = lv[3*b+2];
            const float wbx  = av[3*b], wby  = av[3*b+1], wbz  = av[3*b+2];
            const float mb   = mass[b];
#pragma unroll
            for (int v = 0; v < 8; ++v) {
                if ((t[v] > 0.0f) && !(diagTile && v == vdiag)) {
                    const int m = v + 8 * half;
                    const float dx = bx - pI[m][0];
                    const float dy = by - pI[m][1];
                    const float dz = bz - pI[m][2];
                    const float dist = sqrtf(dx*dx + dy*dy + dz*dz + 1e-12f);
                    const float inv  = 1.0f / (dist + 1e-8f);
                    const float nx = dx*inv, ny = dy*inv, nz = dz*inv;
                    const float ra  = rIr[v];
                    const float pen = (ra + rj) - dist;
                    const float rax = nx*ra, ray = ny*ra, raz = nz*ra;          // r_a
                    const float rbx = rax - dx, rby = ray - dy, rbz = raz - dz; // r_b
                    const float vax = lvI[m][0] + (wI[m][1]*raz - wI[m][2]*ray);
                    const float vay = lvI[m][1] + (wI[m][2]*rax - wI[m][0]*raz);
                    const float vaz = lvI[m][2] + (wI[m][0]*ray - wI[m][1]*rax);
                    const float vbx = lvbx + (wby*rbz - wbz*rby);
                    const float vby = lvby + (wbz*rbx - wbx*rbz);
                    const float vbz = lvbz + (wbx*rby - wby*rbx);
                    const float vn  = (vbx - vax)*nx + (vby - vay)*ny + (vbz - vaz)*nz;
                    if (vn <= 0.0f) {
                        const float ma   = mI[m];
                        const float redm = ma * mb / (ma + mb);
                        const float fm   = -(1.0f + REST) * vn * redm / DT + STIFF * pen;
                        const float fx = -nx * fm, fy = -ny * fm, fz = -nz * fm;
                        atomicAdd(&acc[m*6 + 0], fx);
                        atomicAdd(&acc[m*6 + 1], fy);
                        atomicAdd(&acc[m*6 + 2], fz);
                        atomicAdd(&acc[m*6 + 3], ray*fz - raz*fy);
                        atomicAdd(&acc[m*6 + 4], raz*fx - rax*fz);
                        atomicAdd(&acc[m*6 + 5], rax*fy - ray*fx);
                    }
                }
            }
        }
    }
    __syncthreads();
    if (tid < 96) atomicAdd(&gacc[i0*6 + tid], acc[tid]);
}

// ---------------------------------------------------------------------------
// Kernel 2: semi-implicit Euler integration + quaternion kinematics.
// ---------------------------------------------------------------------------
__global__ void __launch_bounds__(256)
Phys_integrate_kernel(const float* __restrict__ pos,
                      const float* __restrict__ ori,
                      const float* __restrict__ lv,
                      const float* __restrict__ av,
                      const float* __restrict__ rad,
                      const float* __restrict__ mass,
                      const float* __restrict__ extf,
                      const float* __restrict__ gacc,
                      float* __restrict__ out) {
    const int i = blockIdx.x * blockDim.x + threadIdx.x;
    if (i >= NB) return;
    const float m = mass[i];
    // total_force = GRAV*m + external + contact
    const float fx = extf[3*i]   + gacc[i*6 + 0];
    const float fy = extf[3*i+1] + gacc[i*6 + 1];
    const float fz = -9.81f * m + extf[3*i+2] + gacc[i*6 + 2];
    const float vnx = lv[3*i]   + fx / m * DT;
    const float vny = lv[3*i+1] + fy / m * DT;
    const float vnz = lv[3*i+2] + fz / m * DT;
    const float pnx = pos[3*i]   + vnx * DT;
    const float pny = pos[3*i+1] + vny * DT;
    const float pnz = pos[3*i+2] + vnz * DT;
    const float r = rad[i];
    const float invI = 1.0f / (0.4f * m * r * r);
    const float wnx = av[3*i]   + gacc[i*6 + 3] * invI * DT;
    const float wny = av[3*i+1] + gacc[i*6 + 4] * invI * DT;
    const float wnz = av[3*i+2] + gacc[i*6 + 5] * invI * DT;
    // qdot = 0.5 * q (x) (0, w)
    const float qw = ori[4*i], qx = ori[4*i+1], qy = ori[4*i+2], qz = ori[4*i+3];
    const float mw = -qx*wnx - qy*wny - qz*wnz;
    const float mx =  qw*wnx + qy*wnz - qz*wny;
    const float my =  qw*wny - qx*wnz + qz*wnx;
    const float mz =  qw*wnz + qx*wny - qy*wnx;
    float nqw = qw + 0.5f * mw * DT;
    float nqx = qx + 0.5f * mx * DT;
    float nqy = qy + 0.5f * my * DT;
    float nqz = qz + 0.5f * mz * DT;
    const float invn = 1.0f / sqrtf(nqw*nqw + nqx*nqx + nqy*nqy + nqz*nqz);
    nqw *= invn; nqx *= invn; nqy *= invn; nqz *= invn;
    float* o = out + 13 * i;
    o[0] = pnx;  o[1] = pny;  o[2] = pnz;
    o[3] = nqw;  o[4] = nqx;  o[5] = nqy;  o[6] = nqz;
    o[7] = vnx;  o[8] = vny;  o[9] = vnz;
    o[10] = wnx; o[11] = wny; o[12] = wnz;
}

// ---------------------------------------------------------------------------
extern "C" void kernel_launch(void* const* d_in, const int* in_sizes, int n_in,
                              void* d_out, int out_size, void* d_ws, size_t ws_size,
                              hipStream_t stream) {
    const float* pos  = (const float*)d_in[0];
    const float* ori  = (const float*)d_in[1];
    const float* lvel = (const float*)d_in[2];
    const float* avel = (const float*)d_in[3];
    const float* rad  = (const float*)d_in[4];
    const float* mas  = (const float*)d_in[5];
    const float* extf = (const float*)d_in[6];
    float* out  = (float*)d_out;
    float* gacc = (float*)d_ws;              // NB*6 floats = 48 KB scratch

    const int nacc = NB * 6;
    Phys_zero_kernel<<<(nacc + 255) / 256, 256, 0, stream>>>(gacc, nacc);

    dim3 grid(NB / 16, 4);                   // 128 i-tiles x 4 j-chunks
    Phys_collide_kernel<<<grid, 256, 0, stream>>>(pos, lvel, avel, rad, mas, gacc);

    Phys_integrate_kernel<<<(NB + 255) / 256, 256, 0, stream>>>(
        pos, ori, lvel, avel, rad, mas, extf, gacc, out);
}